// Interaction_42374147342439
// MI455X (gfx1250) — compile-verified
//
#include <hip/hip_runtime.h>
#include <math.h>

#define NEDGES 400000
#define NNODES 25000
#define NDIM   256
#define RAD    8
#define MLPW   64
#define NW     256
#define MUL    64

typedef __attribute__((ext_vector_type(2))) float v2f;
typedef __attribute__((ext_vector_type(4))) float v4f;
typedef __attribute__((ext_vector_type(8))) float v8f;

// LDS row strides padded to dodge 64-bank conflicts for the B-fragment /
// A-fragment access patterns (two row-halves offset by 2 rows -> want
// 2*stride mod 64 words to shift banks; 72 -> +16 banks, 264 -> +16 banks,
// 68 -> A-frag rows spread 4 banks apart).
#define W1S 72
#define W2S 72
#define W3S 264
#define HS  68

__device__ __forceinline__ v8f wmma4(v2f a, v2f b, v8f c) {
  // D = A(16x4,f32) * B(4x16,f32) + C(16x16,f32)
  return __builtin_amdgcn_wmma_f32_16x16x4_f32(false, a, false, b, (short)0, c,
                                               false, false);
}

__device__ __forceinline__ float silu_f(float x) {
  // x * sigmoid(x); v_rcp_f32 (1 ulp) instead of IEEE div_scale/fixup chain.
  return x * __builtin_amdgcn_rcpf(1.0f + __expf(-x));
}

__device__ __forceinline__ void atomAddF(float* p, float v) {
  // non-returning global_atomic_add_f32 (tracked on STOREcnt)
  unsafeAtomicAdd(p, v);
}

__global__ void __launch_bounds__(256, 1)
interaction_kernel(const float* __restrict__ fnode,
                   const float* __restrict__ fedge,
                   const float* __restrict__ fw,
                   const int* __restrict__ src,
                   const int* __restrict__ tgt,
                   const float* __restrict__ W1, const float* __restrict__ B1,
                   const float* __restrict__ W2, const float* __restrict__ B2,
                   const float* __restrict__ W3, const float* __restrict__ B3,
                   float* __restrict__ out) {
  __shared__ float sW1[RAD * W1S];
  __shared__ float sW2[MLPW * W2S];
  __shared__ float sW3[MLPW * W3S];
  __shared__ float sB1[MLPW];
  __shared__ float sB2[MLPW];
  __shared__ float sB3[NW];
  __shared__ float sH[8 * 16 * HS];   // per-wave 16x64 staging (stride 68)

  const int tid = threadIdx.x;

  // Cooperative weight staging into LDS (padded strides).
  for (int i = tid; i < RAD * MLPW; i += 256)
    sW1[(i >> 6) * W1S + (i & 63)] = W1[i];
  for (int i = tid; i < MLPW * MLPW; i += 256)
    sW2[(i >> 6) * W2S + (i & 63)] = W2[i];
  for (int i = tid; i < MLPW * NW; i += 256)
    sW3[(i >> 8) * W3S + (i & 255)] = W3[i];
  if (tid < MLPW) { sB1[tid] = B1[tid]; sB2[tid] = B2[tid]; }
  sB3[tid] = B3[tid];  // 256 threads == NW
  __syncthreads();

  const int wv  = tid >> 5;
  const int ln  = tid & 31;
  const int hl  = ln >> 4;   // which half of the wave (K/M offset selector)
  const int l16 = ln & 15;
  float* sh = &sH[wv * 16 * HS];

  const int ntiles  = NEDGES / 16;
  const int wstride = gridDim.x * 8;
  for (int tile = blockIdx.x * 8 + wv; tile < ntiles; tile += wstride) {
    const int e0 = tile * 16;

    // ---------- GEMM1: h1 = silu(fw(16x8) @ W1(8x64) + b1) ----------
    // A-frag (16x4 f32): lane(0..15)->M=l16,K={0,1}; lane(16..31)->K={2,3}
    const float* fwp = fw + (size_t)(e0 + l16) * RAD + 2 * hl;
    v2f a0 = __builtin_nontemporal_load((const v2f*)fwp);        // K-blk 0
    v2f a1 = __builtin_nontemporal_load((const v2f*)(fwp + 4));  // K-blk 1
#pragma unroll
    for (int n = 0; n < 4; ++n) {
      const int col = 16 * n + l16;
      v8f c;
      const float bb = sB1[col];
#pragma unroll
      for (int v = 0; v < 8; ++v) c[v] = bb;
      v2f bk0 = { sW1[(2 * hl + 0) * W1S + col], sW1[(2 * hl + 1) * W1S + col] };
      v2f bk1 = { sW1[(4 + 2 * hl + 0) * W1S + col],
                  sW1[(4 + 2 * hl + 1) * W1S + col] };
      c = wmma4(a0, bk0, c);
      c = wmma4(a1, bk1, c);
      // C/D layout: VGPR v, lane -> row (v + 8*hl), col l16-in-tile
#pragma unroll
      for (int v = 0; v < 8; ++v)
        sh[(v + 8 * hl) * HS + col] = silu_f(c[v]);
    }

    // ---------- GEMM2: h2 = silu(h1(16x64) @ W2(64x64) + b2) ----------
    v8f acc[4];
#pragma unroll
    for (int n = 0; n < 4; ++n) {
      const float bb = sB2[16 * n + l16];
#pragma unroll
      for (int v = 0; v < 8; ++v) acc[n][v] = bb;
    }
    const float* arow = &sh[l16 * HS + 2 * hl];  // A: M=l16, K=4*kk+2*hl+{0,1}
#pragma unroll
    for (int kk = 0; kk < 16; ++kk) {
      v2f a = *(const v2f*)(arow + 4 * kk);      // 8B-aligned ds_load_b64
      const int kr = 4 * kk + 2 * hl;
#pragma unroll
      for (int n = 0; n < 4; ++n) {
        const int col = 16 * n + l16;
        v2f b = { sW2[kr * W2S + col], sW2[(kr + 1) * W2S + col] };
        acc[n] = wmma4(a, b, acc[n]);
      }
    }
#pragma unroll
    for (int n = 0; n < 4; ++n)
#pragma unroll
      for (int v = 0; v < 8; ++v)
        sh[(v + 8 * hl) * HS + 16 * n + l16] = silu_f(acc[n][v]);

    // ---------- per-lane edge metadata: this lane's 8 edge rows ----------
    v4f ed[8]; int tg[8]; int sr[8];
#pragma unroll
    for (int v = 0; v < 8; ++v) {
      const int e = e0 + v + 8 * hl;
      ed[v] = __builtin_nontemporal_load((const v4f*)fedge + e);  // se,ve0..2
      tg[v] = __builtin_nontemporal_load(tgt + e);
      sr[v] = __builtin_nontemporal_load(src + e);
    }

    // Hoist h2 A-fragments into registers: read each once, use 4x in GEMM3.
    v2f afr[16];
#pragma unroll
    for (int kk = 0; kk < 16; ++kk)
      afr[kk] = *(const v2f*)(arow + 4 * kk);

    // ---------- GEMM3 (h2 @ W3 + b3) fused with messages + scatter -------
    const float inv_sqrt3 = 0.57735026918962576f;
    for (int j = 0; j < 4; ++j) {           // channel tile: u = 16j + l16
      const int col = 16 * j + l16;
      v8f wa, wb, wc, wd;
      {
        const float ba = sB3[col], bbv = sB3[64 + col];
        const float bc = sB3[128 + col], bd = sB3[192 + col];
#pragma unroll
        for (int v = 0; v < 8; ++v) { wa[v] = ba; wb[v] = bbv; wc[v] = bc; wd[v] = bd; }
      }
#pragma unroll
      for (int kk = 0; kk < 16; ++kk) {
        const int kr = 4 * kk + 2 * hl;
        v2f bA = { sW3[kr * W3S + col],        sW3[(kr + 1) * W3S + col] };
        v2f bB = { sW3[kr * W3S + 64 + col],   sW3[(kr + 1) * W3S + 64 + col] };
        v2f bC = { sW3[kr * W3S + 128 + col],  sW3[(kr + 1) * W3S + 128 + col] };
        v2f bD = { sW3[kr * W3S + 192 + col],  sW3[(kr + 1) * W3S + 192 + col] };
        wa = wmma4(afr[kk], bA, wa);
        wb = wmma4(afr[kk], bB, wb);
        wc = wmma4(afr[kk], bC, wc);
        wd = wmma4(afr[kk], bD, wd);
      }
      const int u = col;  // channel 0..63
#pragma unroll
      for (int v = 0; v < 8; ++v) {
        const float* np = fnode + (size_t)sr[v] * NDIM;
        const float sn  = np[u];
        const float vn0 = np[MUL + 3 * u + 0];
        const float vn1 = np[MUL + 3 * u + 1];
        const float vn2 = np[MUL + 3 * u + 2];
        const float se = ed[v].x, ve0 = ed[v].y, ve1 = ed[v].z, ve2 = ed[v].w;
        const float dp = vn0 * ve0 + vn1 * ve1 + vn2 * ve2;
        float* op = out + (size_t)tg[v] * 512;
        atomAddF(op + u,        wa[v] * sn * se);
        atomAddF(op + 64 + u,   wb[v] * dp * inv_sqrt3);
        const float wcs = wc[v] * sn;
        atomAddF(op + 128 + 3 * u + 0, wcs * ve0);
        atomAddF(op + 128 + 3 * u + 1, wcs * ve1);
        atomAddF(op + 128 + 3 * u + 2, wcs * ve2);
        const float wds = wd[v] * se;
        atomAddF(op + 320 + 3 * u + 0, wds * vn0);
        atomAddF(op + 320 + 3 * u + 1, wds * vn1);
        atomAddF(op + 320 + 3 * u + 2, wds * vn2);
      }
    }
  }
}

__global__ void zero_kernel(float4* __restrict__ p, int n4) {
  int i = blockIdx.x * blockDim.x + threadIdx.x;
  const int stride = gridDim.x * blockDim.x;
  for (; i < n4; i += stride) p[i] = make_float4(0.f, 0.f, 0.f, 0.f);
}

extern "C" void kernel_launch(void* const* d_in, const int* in_sizes, int n_in,
                              void* d_out, int out_size, void* d_ws, size_t ws_size,
                              hipStream_t stream) {
  const float* fnode = (const float*)d_in[0];
  const float* fedge = (const float*)d_in[1];
  const float* fw    = (const float*)d_in[2];
  const int*   src   = (const int*)d_in[3];
  const int*   tgt   = (const int*)d_in[4];
  const float* W1 = (const float*)d_in[5];
  const float* b1 = (const float*)d_in[6];
  const float* W2 = (const float*)d_in[7];
  const float* b2 = (const float*)d_in[8];
  const float* W3 = (const float*)d_in[9];
  const float* b3 = (const float*)d_in[10];
  float* out = (float*)d_out;

  const int n4 = out_size / 4;  // 25000*512 is divisible by 4
  zero_kernel<<<2048, 256, 0, stream>>>((float4*)out, n4);

  // 25000 edge tiles of 16; 8 waves/block -> 3125 blocks, one tile per wave.
  interaction_kernel<<<3125, 256, 0, stream>>>(fnode, fedge, fw, src, tgt,
                                               W1, b1, W2, b2, W3, b3, out);
}